// TripletLoss_90787018703375
// MI455X (gfx1250) — compile-verified
//
#include <hip/hip_runtime.h>

// Semi-hard triplet loss, N=8192, D=256, fp32, for gfx1250 (MI455X).
// Gram matrix via V_WMMA_F32_16X16X4_F32 (fp32 matrix pipe, keeps reference precision).
// A/B tile staging uses GLOBAL_LOAD_ASYNC_TO_LDS_B128 (ASYNCcnt) when the toolchain
// exposes the builtin; otherwise falls back to load+ds_store.

#define N_PTS   8192
#define DIMS    256
#define MARGIN_F 1.0f
#define BM      64      // anchor rows per workgroup
#define BN      64      // columns per tile
#define KC      64      // K chunk staged in LDS for B
#define F_INF   __builtin_inff()

typedef __attribute__((ext_vector_type(2))) float v2f;
typedef __attribute__((ext_vector_type(8))) float v8f;
typedef __attribute__((ext_vector_type(4))) int   v4i;

#if defined(__gfx1250__) && __has_builtin(__builtin_amdgcn_global_load_async_to_lds_b128)
#define HAVE_ASYNC_LDS 1
#else
#define HAVE_ASYNC_LDS 0
#endif

#if HAVE_ASYNC_LDS
typedef __attribute__((address_space(1))) v4i gv4i;   // global int4
typedef __attribute__((address_space(3))) v4i lv4i;   // LDS int4

static __device__ __forceinline__ gv4i* as_global_v4(const void* p) {
    return (gv4i*)(unsigned long long)(uintptr_t)p;     // AS1 ptr value == generic value
}
static __device__ __forceinline__ lv4i* as_lds_v4(const void* p) {
    return (lv4i*)(unsigned int)(uintptr_t)p;           // low 32 bits = LDS offset
}
static __device__ __forceinline__ void async_wait_all() {
#if __has_builtin(__builtin_amdgcn_s_wait_asynccnt)
    __builtin_amdgcn_s_wait_asynccnt(0);
#endif
}
#endif

// ---------------- Kernel 1: per-row squared norms (wave32 shuffle reduce) ----
__global__ __launch_bounds__(256) void sqnorm_kernel(const float* __restrict__ emb,
                                                     float* __restrict__ sq) {
    int gwave = (blockIdx.x * 256 + threadIdx.x) >> 5;   // one wave per row
    int lane  = threadIdx.x & 31;
    if (gwave >= N_PTS) return;
    const float* row = emb + (size_t)gwave * DIMS;
    float s = 0.0f;
    for (int k = lane; k < DIMS; k += 32) { float v = row[k]; s += v * v; }
    for (int m = 16; m > 0; m >>= 1) s += __shfl_xor(s, m, 32);
    if (lane == 0) sq[gwave] = s;
}

// ---------------- Kernel 2: fused two-pass triplet stats per 64-row block ----
__global__ __launch_bounds__(256) void triplet_kernel(const float* __restrict__ emb,
                                                      const int*   __restrict__ labels,
                                                      const float* __restrict__ sq,
                                                      float* __restrict__ out_loss,
                                                      float* __restrict__ out_valid) {
    __shared__ float As[BM][DIMS + 4];   // 64 x 260 floats (stride 260 -> conflict free frag reads)
    __shared__ float Bs[BN][KC + 4];     // 64 x 68
    __shared__ float Ds[BM][BN + 4];     // dist tile, 64 x 68
    __shared__ float s_sqi[BM], s_sqj[BN];
    __shared__ int   s_li[BM],  s_lj[BN];
    __shared__ float s_dpos[BM], s_dhard[BM], s_dsemi[BM];
    __shared__ float pMax[256], pMin[256];

    const int t    = threadIdx.x;
    const int lane = t & 31;
    const int wave = t >> 5;
    const int wm   = wave & 3;       // 4 wave rows  -> M tile [wm*16, wm*16+16)
    const int wn   = wave >> 2;      // 2 wave cols  -> N strip [wn*32, wn*32+32)
    const int rowbase = blockIdx.x * BM;

    // Stage the A block (64 x 256 fp32) once; reused across all tiles & both passes.
#if HAVE_ASYNC_LDS
    for (int idx = t; idx < BM * (DIMS / 4); idx += 256) {
        int r = idx >> 6;            // DIMS/4 = 64 x b128 per row
        int c = idx & 63;
        const float* gp = emb + (size_t)(rowbase + r) * DIMS + c * 4;
        __builtin_amdgcn_global_load_async_to_lds_b128(as_global_v4(gp),
                                                       as_lds_v4(&As[r][c * 4]), 0, 0);
    }
    async_wait_all();
#else
    {
        const float4* embA = (const float4*)(emb + (size_t)rowbase * DIMS);
        for (int idx = t; idx < BM * (DIMS / 4); idx += 256) {
            int r = idx >> 6;
            int c = idx & 63;
            float4 v = embA[r * (DIMS / 4) + c];
            As[r][c * 4 + 0] = v.x; As[r][c * 4 + 1] = v.y;
            As[r][c * 4 + 2] = v.z; As[r][c * 4 + 3] = v.w;
        }
    }
#endif
    if (t < BM) {
        s_li[t]    = labels[rowbase + t];
        s_sqi[t]   = sq[rowbase + t];
        s_dpos[t]  = -F_INF;
        s_dhard[t] =  F_INF;
        s_dsemi[t] =  F_INF;
    }
    __syncthreads();

    const int mrow  = wm * 16 + (lane & 15);      // A fragment row (M)
    const int khalf = (lane >> 4) << 1;           // upper half-wave holds K+2,K+3

    for (int pass = 0; pass < 2; ++pass) {
        for (int tile = 0; tile < N_PTS / BN; ++tile) {
            const int colbase = tile * BN;
            if (t < BN) {
                s_lj[t]  = labels[colbase + t];
                s_sqj[t] = sq[colbase + t];
            }
            // Prefetch next column tile into the cache hierarchy (global_prefetch_b8).
            if (tile + 1 < N_PTS / BN) {
                const float* pf = emb + (size_t)(colbase + BN + (t & 63)) * DIMS + ((t >> 6) << 6);
                __builtin_prefetch(pf, 0, 1);
            }

            v8f acc0 = {0.f, 0.f, 0.f, 0.f, 0.f, 0.f, 0.f, 0.f};
            v8f acc1 = acc0;

            for (int kc = 0; kc < DIMS; kc += KC) {
                __syncthreads();   // protect Bs (and s_sqj/s_lj on first chunk)
                // Stage B chunk: 64 columns x 64 K values.
#if HAVE_ASYNC_LDS
                for (int idx = t; idx < BN * (KC / 4); idx += 256) {
                    int r = idx >> 4;        // KC/4 = 16 x b128 per column
                    int c = idx & 15;
                    const float* gp = emb + (size_t)(colbase + r) * DIMS + kc + c * 4;
                    __builtin_amdgcn_global_load_async_to_lds_b128(as_global_v4(gp),
                                                                   as_lds_v4(&Bs[r][c * 4]), 0, 0);
                }
                async_wait_all();
#else
                {
                    const float4* embB = (const float4*)emb;
                    for (int idx = t; idx < BN * (KC / 4); idx += 256) {
                        int r = idx >> 4;
                        int c = idx & 15;
                        float4 v = embB[(size_t)(colbase + r) * (DIMS / 4) + (kc >> 2) + c];
                        Bs[r][c * 4 + 0] = v.x; Bs[r][c * 4 + 1] = v.y;
                        Bs[r][c * 4 + 2] = v.z; Bs[r][c * 4 + 3] = v.w;
                    }
                }
#endif
                __syncthreads();

                // fp32 WMMA over the K chunk: 16x16x4 steps, two 16x16 tiles per wave.
                for (int k = 0; k < KC; k += 4) {
                    v2f a  = *(const v2f*)&As[mrow][kc + k + khalf];
                    v2f b0 = *(const v2f*)&Bs[wn * 32 +      (lane & 15)][k + khalf];
                    v2f b1 = *(const v2f*)&Bs[wn * 32 + 16 + (lane & 15)][k + khalf];
                    acc0 = __builtin_amdgcn_wmma_f32_16x16x4_f32(false, a, false, b0,
                                                                 (short)0, acc0, false, false);
                    acc1 = __builtin_amdgcn_wmma_f32_16x16x4_f32(false, a, false, b1,
                                                                 (short)0, acc1, false, false);
                }
            }

            // Unpack accumulators -> L2 distances in LDS tile.
            {
                const int n  = lane & 15;
                const int mh = (lane >> 4) << 3;   // C/D: lanes 16-31 hold M = r+8
                for (int r = 0; r < 8; ++r) {
                    int gm = wm * 16 + r + mh;
                    int gn0 = wn * 32 + n;
                    int gn1 = gn0 + 16;
                    float d2a = s_sqi[gm] + s_sqj[gn0] - 2.0f * acc0[r];
                    float d2b = s_sqi[gm] + s_sqj[gn1] - 2.0f * acc1[r];
                    d2a = fmaxf(d2a, 0.0f);
                    d2b = fmaxf(d2b, 0.0f);
                    Ds[gm][gn0] = (d2a > 0.0f) ? sqrtf(d2a) : 0.0f;
                    Ds[gm][gn1] = (d2b > 0.0f) ? sqrtf(d2b) : 0.0f;
                }
            }
            __syncthreads();

            // Masked row reductions over the 64x64 dist tile: 4 threads per row.
            {
                const int r  = t & 63;
                const int q  = t >> 6;
                const int li = s_li[r];
                const int gi = rowbase + r;
                const float dp = s_dpos[r];        // finalized before pass 1 uses it
                float pmax = -F_INF, pmin = F_INF;
                for (int c = q * 16; c < q * 16 + 16; ++c) {
                    float dv = Ds[r][c];
                    int   lj = s_lj[c];
                    if (pass == 0) {
                        if (li == lj) {
                            if ((colbase + c) != gi) pmax = fmaxf(pmax, dv);
                        } else {
                            pmin = fminf(pmin, dv);
                        }
                    } else {
                        if (li != lj && dv > dp && dv < dp + MARGIN_F)
                            pmin = fminf(pmin, dv);
                    }
                }
                pMax[t] = pmax; pMin[t] = pmin;
            }
            __syncthreads();
            if (t < BM) {
                float mx = fmaxf(fmaxf(pMax[t], pMax[t + 64]), fmaxf(pMax[t + 128], pMax[t + 192]));
                float mn = fminf(fminf(pMin[t], pMin[t + 64]), fminf(pMin[t + 128], pMin[t + 192]));
                if (pass == 0) {
                    s_dpos[t]  = fmaxf(s_dpos[t],  mx);
                    s_dhard[t] = fminf(s_dhard[t], mn);
                } else {
                    s_dsemi[t] = fminf(s_dsemi[t], mn);
                }
            }
            __syncthreads();
        }
    }

    // Per-anchor loss / validity.
    if (t < BM) {
        float dpos  = s_dpos[t];
        float dhard = s_dhard[t];
        float dsemi = s_dsemi[t];
        bool haspos  = dpos  > -F_INF;
        bool hasneg  = dhard <  F_INF;
        bool hassemi = dsemi <  F_INF;
        float dneg = hassemi ? dsemi : dhard;
        float lossv = 0.0f, validv = 0.0f;
        if (haspos && hasneg) {
            validv = 1.0f;
            lossv  = fmaxf(0.0f, dpos - dneg + MARGIN_F);
        }
        out_loss[rowbase + t]  = lossv;
        out_valid[rowbase + t] = validv;
    }
}

// ---------------- Kernel 3: deterministic final reduction ----
__global__ __launch_bounds__(256) void finalize_kernel(const float* __restrict__ loss,
                                                       const float* __restrict__ valid,
                                                       float* __restrict__ out) {
    __shared__ float sl[256], sv[256];
    int t = threadIdx.x;
    float a = 0.0f, b = 0.0f;
    for (int i = t; i < N_PTS; i += 256) { a += loss[i]; b += valid[i]; }
    sl[t] = a; sv[t] = b;
    __syncthreads();
    for (int s = 128; s > 0; s >>= 1) {
        if (t < s) { sl[t] += sl[t + s]; sv[t] += sv[t + s]; }
        __syncthreads();
    }
    if (t == 0) out[0] = sl[0] / sv[0];
}

extern "C" void kernel_launch(void* const* d_in, const int* in_sizes, int n_in,
                              void* d_out, int out_size, void* d_ws, size_t ws_size,
                              hipStream_t stream) {
    (void)in_sizes; (void)n_in; (void)out_size; (void)ws_size;
    const float* emb    = (const float*)d_in[0];
    const int*   labels = (const int*)d_in[1];
    float* ws       = (float*)d_ws;
    float* ws_sq    = ws;                 // [8192]
    float* ws_loss  = ws + N_PTS;         // [8192]
    float* ws_valid = ws + 2 * N_PTS;     // [8192]  -> 96 KB workspace total
    float* outp     = (float*)d_out;

    sqnorm_kernel  <<<N_PTS / 8, 256, 0, stream>>>(emb, ws_sq);
    triplet_kernel <<<N_PTS / BM, 256, 0, stream>>>(emb, labels, ws_sq, ws_loss, ws_valid);
    finalize_kernel<<<1, 256, 0, stream>>>(ws_loss, ws_valid, outp);
}